// VectorQuantiser_4844723110070
// MI455X (gfx1250) — compile-verified
//
#include <hip/hip_runtime.h>
#include <hip/hip_bf16.h>

// ---------------------------------------------------------------------------
// VQ-VAE vector quantiser for MI455X (gfx1250, wave32, WMMA).
//   N = 8*2048 = 16384 input vectors, D = 64, K = 8192 codes.
//   Fused  score = z @ C^T - 0.5*||c||^2  via V_WMMA_F32_16X16X4_F32.
//   K-dimension split across 8 slices -> 8192 waves for SMT latency hiding;
//   per-row winners merged with a packed (score,~idx) atomicMax_u64.
// ---------------------------------------------------------------------------

typedef __attribute__((ext_vector_type(2))) float v2f;
typedef __attribute__((ext_vector_type(8))) float v8f;

#define N_ROWS   16384
#define D_DIM    64
#define K_CODES  8192
#define KSLICES  8
#define BETA_F   0.25f
#define DECAY_F  0.99f
#define EPS_F    1e-05f

// ---- output layout (floats, concatenated in reference return order) -------
#define ZQ_OFF    0                      // 16384*64
#define LOSS_OFF  1048576                // 1
#define IDX_OFF   1048577                // 16384 (as float)
#define NCB_OFF   1064961                // 8192*64
#define NCS_OFF   1589249                // 8192
#define NWS_OFF   1597441                // 8192*64

// ---- workspace layout (floats) --------------------------------------------
#define WS_CS     0                       // [8192]   cluster_size accumulator
#define WS_DW     8192                    // [524288] dw accumulator
#define WS_LOSS   532480                  // [1]      sum (z - zq)^2
#define WS_NTOT   532481                  // [1]      sum(new_cs)
#define WS_CN     532482                  // [8192]   0.5*||c||^2
#define WS_PACK   540674                  // [16384]  u64 packed (score,~idx); 8B aligned
#define WS_IDX    573442                  // [16384]  argmin indices (int32)
#define WS_ZERO_N 532482                  // accumulators to clear per call

// order-preserving float -> uint map (monotone for all finite floats)
__device__ __forceinline__ unsigned int fmap(float f) {
    unsigned int u = __float_as_uint(f);
    return (u & 0x80000000u) ? ~u : (u | 0x80000000u);
}

// ---------------------------------------------------------------------------
__global__ void __launch_bounds__(256)
vq_zero_kernel(float* __restrict__ p, int n) {
    int i = blockIdx.x * 256 + threadIdx.x;
    if (i < n) p[i] = 0.0f;
}

// 0.5*||c||^2 per code row (one thread per code, float4 loads)
__global__ void __launch_bounds__(256)
vq_cnorm_kernel(const float* __restrict__ codebook, float* __restrict__ cn) {
    int k = blockIdx.x * 256 + threadIdx.x;
    const float4* c4 = (const float4*)(codebook + (size_t)k * D_DIM);
    float s = 0.0f;
#pragma unroll
    for (int i = 0; i < 16; ++i) {
        float4 v = c4[i];
        s += v.x * v.x + v.y * v.y + v.z * v.z + v.w * v.w;
    }
    cn[k] = 0.5f * s;
}

// ---------------------------------------------------------------------------
// Fused GEMM + argmax over one K-slice.  One wave32 owns a 16-row A tile and
// scans K/KSLICES codes in 16-wide tiles with V_WMMA_F32_16X16X4_F32.
// A = 16x4 rows of z, B = 4x16 codebook columns, C init = -0.5*||c||^2.
// grid = (N/128 row groups, KSLICES); 8 waves/WG walk the same slice in
// lockstep -> WGP$ reuse of the codebook tiles.
// ---------------------------------------------------------------------------
__global__ void __launch_bounds__(256)
vq_argmin_wmma_kernel(const float* __restrict__ z,
                      const float* __restrict__ codebook,
                      const float* __restrict__ cnormhalf,
                      unsigned long long* __restrict__ packed) {
    const int lane = threadIdx.x & 31;
    const int wave = threadIdx.x >> 5;                  // 0..7
    const int rowBase = (blockIdx.x * 8 + wave) * 16;   // 16 rows per wave
    const int m = lane & 15;   // A: row-in-tile / B: code-col-in-tile
    const int h = lane >> 4;   // selects K pair {2h, 2h+1} per the ISA layout
    const int kt0 = blockIdx.y * (K_CODES / 16 / KSLICES);   // 64 tiles/slice

    // ---- preload A fragments for all 16 K-chunks (stay in VGPRs) -----------
    v2f a[16];
    const float* zrow = z + (size_t)(rowBase + m) * D_DIM + 2 * h;
#pragma unroll
    for (int c = 0; c < 16; ++c)
        a[c] = *(const v2f*)(zrow + 4 * c);

    float best[8];
    int   bidx[8];
#pragma unroll
    for (int r = 0; r < 8; ++r) { best[r] = -3.4e38f; bidx[r] = 0; }

    // ---- sweep this slice's 64 code tiles ---------------------------------
    for (int kt = kt0; kt < kt0 + (K_CODES / 16 / KSLICES); ++kt) {
        const int code = kt * 16 + m;
        const float* crow = codebook + (size_t)code * D_DIM + 2 * h;
        const float negc = -cnormhalf[code];

        // stage the whole B tile first so the 16 b64 loads clause together
        v2f b[16];
#pragma unroll
        for (int c = 0; c < 16; ++c)
            b[c] = *(const v2f*)(crow + 4 * c);

        v8f acc;
#pragma unroll
        for (int r = 0; r < 8; ++r) acc[r] = negc;   // C = -0.5*||c||^2

#pragma unroll
        for (int c = 0; c < 16; ++c)
            acc = __builtin_amdgcn_wmma_f32_16x16x4_f32(
                      false, a[c], false, b[c], (short)0, acc, false, false);

        // running argmax; kt increases => strict '>' keeps the lowest index
#pragma unroll
        for (int r = 0; r < 8; ++r) {
            if (acc[r] > best[r]) { best[r] = acc[r]; bidx[r] = code; }
        }
    }

    // ---- reduce across the 16 code-columns, then merge slices via atomic ---
#pragma unroll
    for (int r = 0; r < 8; ++r) {
        float v = best[r];
        int   iv = bidx[r];
        for (int mm = 1; mm < 16; mm <<= 1) {
            float ov = __shfl_xor(v, mm, 16);
            int   oi = __shfl_xor(iv, mm, 16);
            if (ov > v || (ov == v && oi < iv)) { v = ov; iv = oi; }
        }
        if (m == 0) {
            int row = rowBase + r + 8 * h;   // C/D layout: M = r + 8*(lane>=16)
            // high word: monotone-mapped score; low word: 0xFFFFFFFF - idx so
            // atomicMax prefers the LOWER index on equal scores (argmin ties).
            unsigned long long pk =
                ((unsigned long long)fmap(v) << 32) |
                (unsigned long long)(0xFFFFFFFFu - (unsigned int)iv);
            atomicMax(&packed[row], pk);
        }
    }
}

// unpack per-row winners
__global__ void __launch_bounds__(256)
vq_extract_kernel(const unsigned long long* __restrict__ packed,
                  int* __restrict__ idx_i32,
                  float* __restrict__ idx_f32_out) {
    int row = blockIdx.x * 256 + threadIdx.x;
    unsigned long long pk = packed[row];
    int iv = (int)(0xFFFFFFFFu - (unsigned int)(pk & 0xFFFFFFFFull));
    idx_i32[row] = iv;
    idx_f32_out[row] = (float)iv;
}

// ---------------------------------------------------------------------------
// Gather z_q (= z_q_st numerically), loss partial sums, scatter-add dw / cs.
// 256 threads = 4 rows x 64 dims.
// ---------------------------------------------------------------------------
__global__ void __launch_bounds__(256)
vq_gather_scatter_kernel(const float* __restrict__ z,
                         const float* __restrict__ codebook,
                         const int*   __restrict__ idx,
                         float* __restrict__ zq_out,
                         float* __restrict__ dw,
                         float* __restrict__ cs,
                         float* __restrict__ loss_accum) {
    __shared__ float sm[256];
    const int t   = threadIdx.x;
    const int row = blockIdx.x * 4 + (t >> 6);
    const int d   = t & 63;

    const int id  = idx[row];
    const float c  = codebook[(size_t)id * D_DIM + d];
    const float zv = z[(size_t)row * D_DIM + d];
    zq_out[(size_t)row * D_DIM + d] = c;

    const float diff = zv - c;
    sm[t] = diff * diff;

    atomicAdd(&dw[(size_t)id * D_DIM + d], zv);
    if (d == 0) atomicAdd(&cs[id], 1.0f);

    __syncthreads();
#pragma unroll
    for (int s = 128; s > 0; s >>= 1) {
        if (t < s) sm[t] += sm[t + s];
        __syncthreads();
    }
    if (t == 0) atomicAdd(loss_accum, sm[0]);
}

// new_cs = ema*decay + cs*(1-decay); also accumulate n = sum(new_cs)
__global__ void __launch_bounds__(256)
vq_newcs_kernel(const float* __restrict__ ema_cs,
                const float* __restrict__ cs,
                float* __restrict__ newcs_out,
                float* __restrict__ ntot) {
    __shared__ float sm[256];
    const int t = threadIdx.x;
    const int k = blockIdx.x * 256 + t;
    const float v = ema_cs[k] * DECAY_F + cs[k] * (1.0f - DECAY_F);
    newcs_out[k] = v;
    sm[t] = v;
    __syncthreads();
#pragma unroll
    for (int s = 128; s > 0; s >>= 1) {
        if (t < s) sm[t] += sm[t + s];
        __syncthreads();
    }
    if (t == 0) atomicAdd(ntot, sm[0]);
}

// new_ws, new_codebook, vq_loss
__global__ void __launch_bounds__(256)
vq_finalize_kernel(const float* __restrict__ ema_ws,
                   const float* __restrict__ dw,
                   const float* __restrict__ newcs,
                   const float* __restrict__ ntot,
                   const float* __restrict__ loss_sum,
                   float* __restrict__ newws_out,
                   float* __restrict__ newcb_out,
                   float* __restrict__ vqloss_out) {
    const size_t i = (size_t)blockIdx.x * 256 + threadIdx.x;   // over K*D
    const int k = (int)(i >> 6);
    const float nws = ema_ws[i] * DECAY_F + dw[i] * (1.0f - DECAY_F);
    newws_out[i] = nws;
    const float n = *ntot;
    const float ncs = newcs[k];
    const float smoothed = (ncs + EPS_F) / (n + (float)K_CODES * EPS_F) * n;
    newcb_out[i] = nws / smoothed;
    if (i == 0)
        *vqloss_out = (1.0f + BETA_F) * (*loss_sum) / (float)(N_ROWS * D_DIM);
}

// ---------------------------------------------------------------------------
extern "C" void kernel_launch(void* const* d_in, const int* in_sizes, int n_in,
                              void* d_out, int out_size, void* d_ws, size_t ws_size,
                              hipStream_t stream) {
    const float* z        = (const float*)d_in[0];
    const float* codebook = (const float*)d_in[1];
    const float* ema_cs   = (const float*)d_in[2];
    const float* ema_ws   = (const float*)d_in[3];

    float* out = (float*)d_out;
    float* ws  = (float*)d_ws;

    float* cs    = ws + WS_CS;
    float* dw    = ws + WS_DW;
    float* loss  = ws + WS_LOSS;
    float* ntot  = ws + WS_NTOT;
    float* cn    = ws + WS_CN;
    unsigned long long* packed = (unsigned long long*)(ws + WS_PACK);
    int*   idx   = (int*)(ws + WS_IDX);

    // 1) clear accumulators (cs, dw, loss, ntot) and packed winners
    vq_zero_kernel<<<(WS_ZERO_N + 255) / 256, 256, 0, stream>>>(ws, WS_ZERO_N);
    vq_zero_kernel<<<(2 * N_ROWS + 255) / 256, 256, 0, stream>>>(
        ws + WS_PACK, 2 * N_ROWS);

    // 2) 0.5*||c||^2
    vq_cnorm_kernel<<<K_CODES / 256, 256, 0, stream>>>(codebook, cn);

    // 3) fused WMMA distance + argmin
    //    grid: 128 row-groups x 8 K-slices, 8 waves/WG -> 8192 waves
    dim3 agrid(N_ROWS / 128, KSLICES, 1);
    vq_argmin_wmma_kernel<<<agrid, 256, 0, stream>>>(z, codebook, cn, packed);

    // 3b) unpack winners
    vq_extract_kernel<<<N_ROWS / 256, 256, 0, stream>>>(
        packed, idx, out + IDX_OFF);

    // 4) gather z_q, loss partials, scatter-add cs/dw
    vq_gather_scatter_kernel<<<N_ROWS / 4, 256, 0, stream>>>(
        z, codebook, idx, out + ZQ_OFF, dw, cs, loss);

    // 5) new_cs + n
    vq_newcs_kernel<<<K_CODES / 256, 256, 0, stream>>>(
        ema_cs, cs, out + NCS_OFF, ntot);

    // 6) new_ws, new_codebook, vq_loss
    vq_finalize_kernel<<<(K_CODES * D_DIM) / 256, 256, 0, stream>>>(
        ema_ws, dw, out + NCS_OFF, ntot, loss,
        out + NWS_OFF, out + NCB_OFF, out + LOSS_OFF);
}